// Encoder_90632399880827
// MI455X (gfx1250) — compile-verified
//
#include <hip/hip_runtime.h>

typedef float v2f __attribute__((ext_vector_type(2)));
typedef float v8f __attribute__((ext_vector_type(8)));
typedef int   v4i __attribute__((ext_vector_type(4)));
typedef __attribute__((address_space(1))) v4i* gv4i_ptr;   // global v4i*
typedef __attribute__((address_space(3))) v4i* lv4i_ptr;   // LDS v4i*

#define NJ 24              // joints per frame
#define HIDC 24            // hidden channels (3 heads x 8)
#define NPW 48             // nodes per wave (2 frames)
#define XSTR 25            // x buffer row stride (gcd(25,64)=1, conflict-free)
#define HSTR 33            // h buffer row stride: room for N-tile cols 16..31,
                           // unconditional WMMA write-back, gcd(33,64)=1
#define FRAMES_PER_BLOCK 8 // 4 waves * 2 frames

__device__ __constant__ int kPar[NJ] = {
    -1, 0, 0, 0, 1, 2, 3, 4, 5, 6, 7, 8, 9, 9, 9, 12, 13, 14, 16, 17, 18, 19, 20, 21};

__global__ __launch_bounds__(128) void gat_smpl_kernel(
    const float* __restrict__ src,   // (frames, 72)
    const float* __restrict__ Wpre,  // (3, 24)
    const float* __restrict__ bpre,  // (24)
    const float* __restrict__ Wg,    // (24, 24)
    const float* __restrict__ attS,  // (3, 8)
    const float* __restrict__ attD,  // (3, 8)
    const float* __restrict__ bg,    // (24)
    float* __restrict__ outg,        // (frames, 576)
    int nframes)
{
    __shared__ float sWpre[72], sBpre[24], sWg[576], sAS[24], sAD[24], sBg[24];
    __shared__ __align__(16) float sX[4][NPW * XSTR];  // x; reused as PACKED out staging
    __shared__ __align__(16) float sH[4][NPW * HSTR];  // h (cols 24..31 scratch)
    __shared__ float sA[4][NPW * 6];                   // per node: a_src[3], a_dst[3]

    const int tid  = threadIdx.x;
    const int wave = tid >> 5;
    const int lane = tid & 31;

    // ---- load weights into LDS (broadcast-friendly) ----
    for (int i = tid; i < 72;  i += 128) sWpre[i] = Wpre[i];
    for (int i = tid; i < 576; i += 128) sWg[i]   = Wg[i];
    if (tid < 24) {
        sBpre[tid] = bpre[tid];
        sAS[tid]   = attS[tid];
        sAD[tid]   = attD[tid];
        sBg[tid]   = bg[tid];
    }
    __syncthreads();

    const long frame0 = (long)blockIdx.x * FRAMES_PER_BLOCK + wave * 2;
    const bool wvalid = (frame0 + 2) <= (long)nframes;
    const float* srcW = src  + frame0 * (NJ * 3);
    float*       outW = outg + frame0 * (NJ * HIDC);
    float* xw = sX[wave];
    float* hw = sH[wave];
    float* aw = sA[wave];

    // ---- stage 1: x = relu(src @ Wpre + bpre), one node per lane ----
    #pragma unroll
    for (int t = 0; t < 2; ++t) {
        int node = lane + 32 * t;
        if (wvalid && node < NPW) {
            const float* p = srcW + node * 3;   // contiguous across lanes -> coalesced
            float x0 = __builtin_nontemporal_load(p);
            float x1 = __builtin_nontemporal_load(p + 1);
            float x2 = __builtin_nontemporal_load(p + 2);
            #pragma unroll
            for (int c = 0; c < HIDC; ++c) {
                float v = fmaf(x0, sWpre[c],
                          fmaf(x1, sWpre[24 + c],
                          fmaf(x2, sWpre[48 + c], sBpre[c])));
                xw[node * XSTR + c] = fmaxf(v, 0.0f);
            }
        }
    }
    __syncthreads();

    // ---- stage 2: h = x @ Wg via V_WMMA_F32_16X16X4_F32 ----
    // A (16x4 f32): lanes 0-15 rows M; VGPR0 = K,K+2 halves; VGPR1 = K+1,K+3
    // B (4x16 f32): lanes = N columns; half-lane groups carry K, K+2
    const int half = lane >> 4;   // 0|1
    const int l15  = lane & 15;

    v2f bf[2][6];                 // [N-tile][K-step] B fragments, reused by all M-tiles
    #pragma unroll
    for (int nt = 0; nt < 2; ++nt) {
        int col = l15 + nt * 16;
        bool ok = col < HIDC;     // zero-pad columns 24..31
        #pragma unroll
        for (int ks = 0; ks < 6; ++ks) {
            int k0 = ks * 4 + half * 2;
            bf[nt][ks].x = ok ? sWg[k0 * HIDC + col] : 0.0f;
            bf[nt][ks].y = ok ? sWg[(k0 + 1) * HIDC + col] : 0.0f;
        }
    }

    #pragma unroll
    for (int mt = 0; mt < 3; ++mt) {      // 48 rows = 3 tiles of 16
        v8f acc0 = {0.f,0.f,0.f,0.f,0.f,0.f,0.f,0.f};
        v8f acc1 = {0.f,0.f,0.f,0.f,0.f,0.f,0.f,0.f};
        #pragma unroll
        for (int ks = 0; ks < 6; ++ks) {  // K = 24 in 6 steps of 4
            int k0 = ks * 4 + half * 2;
            v2f a;
            a.x = xw[(mt * 16 + l15) * XSTR + k0];
            a.y = xw[(mt * 16 + l15) * XSTR + k0 + 1];
            acc0 = __builtin_amdgcn_wmma_f32_16x16x4_f32(
                false, a, false, bf[0][ks], (short)0, acc0, false, false);
            acc1 = __builtin_amdgcn_wmma_f32_16x16x4_f32(
                false, a, false, bf[1][ks], (short)0, acc1, false, false);
        }
        // C/D layout: VGPR r, lanes 0-15 -> M=r, lanes 16-31 -> M=8+r; N = lane&15
        // Unconditional stores: cols 24..31 land in the HSTR scratch pad.
        #pragma unroll
        for (int r = 0; r < 8; ++r) {
            int node = mt * 16 + half * 8 + r;
            hw[node * HSTR + l15]      = acc0[r];
            hw[node * HSTR + 16 + l15] = acc1[r];
        }
    }
    __syncthreads();

    // ---- stage 3: attention logits a_src/a_dst per node/head ----
    #pragma unroll
    for (int t = 0; t < 2; ++t) {
        int node = lane + 32 * t;
        if (node < NPW) {
            #pragma unroll
            for (int hh = 0; hh < 3; ++hh) {
                float as = 0.f, ad = 0.f;
                #pragma unroll
                for (int o = 0; o < 8; ++o) {
                    float hv = hw[node * HSTR + hh * 8 + o];
                    as = fmaf(hv, sAS[hh * 8 + o], as);
                    ad = fmaf(hv, sAD[hh * 8 + o], ad);
                }
                aw[node * 6 + hh]     = as;
                aw[node * 6 + 3 + hh] = ad;
            }
        }
    }
    __syncthreads();

    // ---- stage 4: 2-way softmax (self + parent) + aggregate + bias + relu ----
    // Result written PACKED (stride 24) into xw for contiguous async store-out.
    #pragma unroll
    for (int t = 0; t < 2; ++t) {
        int node = lane + 32 * t;
        if (node < NPW) {
            int f = node / NJ, j = node - f * NJ;
            int p = kPar[j];
            int par = (p >= 0) ? (f * NJ + p) : node;
            float alS[3], alP[3];
            #pragma unroll
            for (int hh = 0; hh < 3; ++hh) {
                float es = aw[node * 6 + hh] + aw[node * 6 + 3 + hh];
                es = es > 0.f ? es : 0.2f * es;              // leaky_relu(0.2)
                if (p >= 0) {
                    float ep = aw[par * 6 + hh] + aw[node * 6 + 3 + hh];
                    ep = ep > 0.f ? ep : 0.2f * ep;
                    float mm = fmaxf(es, ep);
                    float xs = __expf(es - mm), xp = __expf(ep - mm);
                    float inv = 1.0f / (xs + xp);
                    alS[hh] = xs * inv; alP[hh] = xp * inv;
                } else {
                    alS[hh] = 1.0f; alP[hh] = 0.0f;          // root: self-loop only
                }
            }
            #pragma unroll
            for (int c = 0; c < HIDC; ++c) {
                int hh = c >> 3;
                float v = alS[hh] * hw[node * HSTR + c]
                        + alP[hh] * hw[par * HSTR + c] + sBg[c];
                xw[node * HIDC + c] = fmaxf(v, 0.0f);        // packed staging
            }
        }
    }
    __syncthreads();   // ds writes complete before LDS is read (by us or by async DMA)

    // ---- stream out: 4608 contiguous bytes per wave ----
    if (wvalid) {
#if defined(__gfx1250__) && __has_builtin(__builtin_amdgcn_global_store_async_from_lds_b128)
        // CDNA5 async DMA path: 16 B per lane per op, no VGPR round-trip,
        // tracked by ASYNCcnt; S_ENDPGM's implicit wait-idle covers completion.
        #pragma unroll
        for (int it = 0; it < (NPW * HIDC * 4) / (32 * 16); ++it) {   // 9 ops
            int byteoff = (lane + 32 * it) * 16;
            __builtin_amdgcn_global_store_async_from_lds_b128(
                (gv4i_ptr)(void*)((char*)outW + byteoff),
                (lv4i_ptr)(void*)((char*)xw + byteoff),
                0, 0);
        }
#else
        // Fallback: coalesced non-temporal stores through VGPRs.
        for (int idx = lane; idx < NPW * HIDC; idx += 32) {
            __builtin_nontemporal_store(xw[idx], outW + idx);
        }
#endif
    }
}

extern "C" void kernel_launch(void* const* d_in, const int* in_sizes, int n_in,
                              void* d_out, int out_size, void* d_ws, size_t ws_size,
                              hipStream_t stream) {
    const float* src  = (const float*)d_in[0];
    const float* Wpre = (const float*)d_in[1];
    const float* bpre = (const float*)d_in[2];
    const float* Wg   = (const float*)d_in[3];
    const float* attS = (const float*)d_in[4];
    const float* attD = (const float*)d_in[5];
    const float* bg   = (const float*)d_in[6];
    float* outg = (float*)d_out;

    const int nframes = in_sizes[0] / (NJ * 3);              // 64*2048 = 131072
    const int blocks  = (nframes + FRAMES_PER_BLOCK - 1) / FRAMES_PER_BLOCK;

    gat_smpl_kernel<<<blocks, 128, 0, stream>>>(src, Wpre, bpre, Wg, attS, attD, bg,
                                                outg, nframes);
}